// Gridding_71090298683883
// MI455X (gfx1250) — compile-verified
//
#include <hip/hip_runtime.h>

#define GN1 256
#define GN2 256
#define GN3 256

// ---------------------------------------------------------------------------
// Trilinear gridding scatter: one point per lane (wave32), 8 hw f32 atomics.
// Inputs are streamed with non-temporal hints so they don't evict the
// L2-resident 64MB volume. Placed FIRST in the file so the disasm snippet
// shows the atomic lowering (expect global_atomic_add_f32, no return).
// ---------------------------------------------------------------------------
__global__ __launch_bounds__(256) void gridding_scatter_kernel(
    const float* __restrict__ rho,
    const float* __restrict__ xs,
    float* __restrict__ vol,
    int m) {
    int i = blockIdx.x * blockDim.x + threadIdx.x;
    if (i >= m) return;

    // Non-temporal streaming loads (TH=NT): 32MB read exactly once.
    float r = __builtin_nontemporal_load(rho + i);
    float x = __builtin_nontemporal_load(xs + 3 * i + 0);
    float y = __builtin_nontemporal_load(xs + 3 * i + 1);
    float z = __builtin_nontemporal_load(xs + 3 * i + 2);

    // c = (x - p/2)/p = x*256 - 0.5  (p = 1/256)
    float cx = __builtin_fmaf(x, 256.0f, -0.5f);
    float cy = __builtin_fmaf(y, 256.0f, -0.5f);
    float cz = __builtin_fmaf(z, 256.0f, -0.5f);

    float fx = floorf(cx), fy = floorf(cy), fz = floorf(cz);
    float dx = cx - fx,    dy = cy - fy,    dz = cz - fz;
    int bx = (int)fx, by = (int)fy, bz = (int)fz;

    // Fold rho into the x-axis weights.
    float wx[2] = { r * (1.0f - dx), r * dx };
    float wy[2] = { 1.0f - dy, dy };
    float wz[2] = { 1.0f - dz, dz };

#pragma unroll
    for (int ii = 0; ii < 2; ++ii) {
        int ix = bx + ii;
        if (ix < 0 || ix >= GN1) continue;
#pragma unroll
        for (int jj = 0; jj < 2; ++jj) {
            int iy = by + jj;
            if (iy < 0 || iy >= GN2) continue;
            float wxy = wx[ii] * wy[jj];
            int rowbase = (ix * GN2 + iy) * GN3;
#pragma unroll
            for (int kk = 0; kk < 2; ++kk) {
                int iz = bz + kk;
                if (iz < 0 || iz >= GN3) continue;
                // Non-returning hw f32 atomic add, device (agent) scope ->
                // global_atomic_add_f32 served in L2 (volume is L2-resident).
                (void)__hip_atomic_fetch_add(vol + rowbase + iz,
                                             wxy * wz[kk],
                                             __ATOMIC_RELAXED,
                                             __HIP_MEMORY_SCOPE_AGENT);
            }
        }
    }
}

// ---------------------------------------------------------------------------
// Zero the 256^3 volume with wide stores (also pulls/dirties the 64MB volume
// into the 192MB L2 right before the atomic storm hits it).
// ---------------------------------------------------------------------------
__global__ __launch_bounds__(256) void zero_vol_kernel(float4* __restrict__ out, int n4) {
    int i = blockIdx.x * blockDim.x + threadIdx.x;
    if (i < n4) {
        out[i] = make_float4(0.f, 0.f, 0.f, 0.f);
    }
}

extern "C" void kernel_launch(void* const* d_in, const int* in_sizes, int n_in,
                              void* d_out, int out_size, void* d_ws, size_t ws_size,
                              hipStream_t stream) {
    const float* rho = (const float*)d_in[0];   // [M] float32
    const float* xs  = (const float*)d_in[1];   // [M,3] float32
    float* vol = (float*)d_out;                 // [256*256*256] float32
    int m = in_sizes[0];

    // 1) Zero the volume (harness poisons d_out; must re-zero every call).
    int n4 = out_size / 4;  // out_size = 256^3 floats, divisible by 4
    int zblocks = (n4 + 255) / 256;
    zero_vol_kernel<<<zblocks, 256, 0, stream>>>((float4*)vol, n4);

    // 2) Scatter: one point per lane.
    int gblocks = (m + 255) / 256;
    gridding_scatter_kernel<<<gblocks, 256, 0, stream>>>(rho, xs, vol, m);
}